// DiffEncoder_62534723830360
// MI455X (gfx1250) — compile-verified
//
#include <hip/hip_runtime.h>

typedef __attribute__((ext_vector_type(2))) float v2f;
typedef __attribute__((ext_vector_type(8))) float v8f;

#define SEQ_LEN 262144
#define NGATES  12      // 3 * STATE_DIM
#define HDIM    4

__device__ __forceinline__ float bcast_lane(float v, int lane) {
  return __int_as_float(__builtin_amdgcn_readlane(__float_as_int(v), lane));
}

// sigmoid(x) = 1 / (1 + 2^(-log2(e)*x)), fast rcp (~1 ulp)
__device__ __forceinline__ float fast_sigmoid(float x) {
  const float e = __builtin_amdgcn_exp2f(-1.44269504088896340736f * x);
  return __builtin_amdgcn_rcpf(1.0f + e);
}

// ---------------------------------------------------------------------------
// Kernel 1: gi[gate, t] = x[t,:] @ w_ih.T + b_ih (SoA layout), via
// V_WMMA_F32_16X16X4_F32. One wave = one 16-timestep tile.
//   A (16x4): row m = timestep t0+m, K = {x0, x1, 0, 0}
//   B (4x16): col n = gate (n<12),   K = {w_ih[n][0], w_ih[n][1], 0, 0}
//   C: b_ih[n] broadcast down each column.
// SoA output lets each lane store its 8 results as two b128 stores.
// ---------------------------------------------------------------------------
__global__ __launch_bounds__(256) void gi_wmma_kernel(
    const float* __restrict__ x, const float* __restrict__ w_ih,
    const float* __restrict__ b_ih, float* __restrict__ gi) {
  const int lane = threadIdx.x & 31;
  const int wave = threadIdx.x >> 5;
  const int tile = blockIdx.x * 8 + wave;
  const int t0   = tile * 16;
  const int half = lane >> 4;   // 0 -> K={0,1} ; 1 -> K={2,3}
  const int idx  = lane & 15;

  v2f a;
  if (half == 0) {
    const float2 xv = *reinterpret_cast<const float2*>(x + (size_t)(t0 + idx) * 2);
    a.x = xv.x; a.y = xv.y;
  } else {
    a.x = 0.0f; a.y = 0.0f;
  }

  v2f b;
  if (half == 0 && idx < NGATES) {
    const float2 wv = *reinterpret_cast<const float2*>(w_ih + idx * 2);
    b.x = wv.x; b.y = wv.y;
  } else {
    b.x = 0.0f; b.y = 0.0f;
  }

  const float bias = (idx < NGATES) ? b_ih[idx] : 0.0f;
  v8f c;
#pragma unroll
  for (int i = 0; i < 8; ++i) c[i] = bias;

  v8f d = __builtin_amdgcn_wmma_f32_16x16x4_f32(
      /*neg_a=*/false, a, /*neg_b=*/false, b,
      /*c_mod=*/(short)0, c, /*reuse_a=*/false, /*reuse_b=*/false);

  // D: lane owns column idx; rows (half*8)..(half*8+7) -> contiguous in SoA
  if (idx < NGATES) {
    float* dst = gi + (size_t)idx * SEQ_LEN + t0 + half * 8;
    *reinterpret_cast<float4*>(dst + 0) = make_float4(d[0], d[1], d[2], d[3]);
    *reinterpret_cast<float4*>(dst + 4) = make_float4(d[4], d[5], d[6], d[7]);
  }
}

// ---------------------------------------------------------------------------
// Kernel 2: serial GRU scan, one wave32. Lane j (0-11) owns gate j:
// lanes 0-3 r, 4-7 z, 8-11 n. Dependent chain per step:
//   3-deep FMA tree (gh) -> exp+rcp (sigmoid) -> bpermute r -> exp+rcp (tanh)
//   -> bpermute z,n -> 1 FMA -> 4 readlane broadcasts.
// gi prefetched 8 steps ahead in registers, unconditionally (overrun lands in
// the allocated hbuf region and is never consumed); lanes >= 12 alias gate 0.
// ---------------------------------------------------------------------------
__global__ __launch_bounds__(32) void gru_scan_kernel(
    const float* __restrict__ gi, const float* __restrict__ state0,
    const float* __restrict__ w_hh, const float* __restrict__ b_hh,
    float* __restrict__ hbuf, float* __restrict__ state_out) {
  const int lane = threadIdx.x;            // 0..31
  const int j    = (lane < NGATES) ? lane : 0;
  const int jj   = lane & 3;

  const float w0 = w_hh[j * 4 + 0];
  const float w1 = w_hh[j * 4 + 1];
  const float w2 = w_hh[j * 4 + 2];
  const float w3 = w_hh[j * 4 + 3];
  const float bh = b_hh[j];

  float h0 = state0[0], h1 = state0[1], h2 = state0[2], h3 = state0[3];

  const float* __restrict__ gptr = gi + (size_t)j * SEQ_LEN;  // SoA row

  constexpr int PF = 8;   // prefetch depth (register pipeline)
  float buf[PF];
#pragma unroll
  for (int u = 0; u < PF; ++u) buf[u] = gptr[u];

  for (int t = 0; t < SEQ_LEN; t += PF) {
#pragma unroll
    for (int u = 0; u < PF; ++u) {
      const float giv = buf[u];
      buf[u] = gptr[t + u + PF];   // unconditional prefetch, 8 steps ahead

      // gh_j = b_hh[j] + h . w_hh[j,:]  (depth-3 tree)
      const float p01 = fmaf(h1, w1, fmaf(h0, w0, bh));
      const float p23 = fmaf(h3, w3, h2 * w2);
      const float gh  = p01 + p23;

      // lanes 0-7: sigmoid(gi + gh)  (0-3 -> r, 4-7 -> z)
      const float s = fast_sigmoid(giv + gh);

      // lanes 8-11 gather r_{jj} from lane jj
      const float r = __shfl(s, jj, 32);

      // lanes 8-11: n = tanh(gi_n + r*gh_n) = 2*sigmoid(2x) - 1
      const float nv = fmaf(2.0f, fast_sigmoid(2.0f * (giv + r * gh)), -1.0f);

      // each lane updates component jj: h' = n + z*(h - n)
      const float z    = __shfl(s, 4 + jj, 32);
      const float nn   = __shfl(nv, 8 + jj, 32);
      const float hold = (jj == 0) ? h0 : (jj == 1) ? h1 : (jj == 2) ? h2 : h3;
      const float hnew = fmaf(z, hold - nn, nn);

      // stream h[t] (lanes 0-3 hold the true components); off the chain
      if (lane < HDIM) hbuf[(size_t)(t + u) * HDIM + lane] = hnew;

      // broadcast new state via v_readlane (no LDS round-trip)
      h0 = bcast_lane(hnew, 0);
      h1 = bcast_lane(hnew, 1);
      h2 = bcast_lane(hnew, 2);
      h3 = bcast_lane(hnew, 3);
    }
  }

  if (lane == 0) {
    state_out[0] = h0; state_out[1] = h1;
    state_out[2] = h2; state_out[3] = h3;
  }
}

// ---------------------------------------------------------------------------
// Kernel 3: y[t, 0:2] = h[t, :] @ w_out.T + b_out via WMMA f32 16x16x4.
//   A (16x4): row m = timestep, K = h components 0..3
//   B (4x16): col n < 2, K = w_out[n][0..3] ;  C: b_out[n] per column.
// ---------------------------------------------------------------------------
__global__ __launch_bounds__(256) void yproj_wmma_kernel(
    const float* __restrict__ hbuf, const float* __restrict__ w_out,
    const float* __restrict__ b_out, float* __restrict__ y) {
  const int lane = threadIdx.x & 31;
  const int wave = threadIdx.x >> 5;
  const int tile = blockIdx.x * 8 + wave;
  const int t0   = tile * 16;
  const int half = lane >> 4;
  const int idx  = lane & 15;

  v2f a;
  {
    const float2 hv = *reinterpret_cast<const float2*>(
        hbuf + (size_t)(t0 + idx) * 4 + half * 2);
    a.x = hv.x; a.y = hv.y;
  }

  v2f b;
  if (idx < 2) {
    const float2 wv = *reinterpret_cast<const float2*>(w_out + idx * 4 + half * 2);
    b.x = wv.x; b.y = wv.y;
  } else {
    b.x = 0.0f; b.y = 0.0f;
  }

  const float bias = (idx < 2) ? b_out[idx] : 0.0f;
  v8f c;
#pragma unroll
  for (int i = 0; i < 8; ++i) c[i] = bias;

  v8f d = __builtin_amdgcn_wmma_f32_16x16x4_f32(
      false, a, false, b, (short)0, c, false, false);

  if (idx < 2) {
    const int rbase = half * 8;
#pragma unroll
    for (int i = 0; i < 8; ++i)
      y[(size_t)(t0 + rbase + i) * 2 + idx] = d[i];
  }
}

// ---------------------------------------------------------------------------
extern "C" void kernel_launch(void* const* d_in, const int* in_sizes, int n_in,
                              void* d_out, int out_size, void* d_ws, size_t ws_size,
                              hipStream_t stream) {
  const float* x      = (const float*)d_in[0];  // (SEQ_LEN, 1, 2)
  const float* state  = (const float*)d_in[1];  // (1, 4)
  const float* w_ih   = (const float*)d_in[2];  // (12, 2)
  const float* w_hh   = (const float*)d_in[3];  // (12, 4)
  const float* b_ih   = (const float*)d_in[4];  // (12,)
  const float* b_hh   = (const float*)d_in[5];  // (12,)
  const float* w_out  = (const float*)d_in[6];  // (2, 4)
  const float* b_out  = (const float*)d_in[7];  // (2,)

  float* out  = (float*)d_out;                    // y: SEQ_LEN*2, then state: 4
  float* gi   = (float*)d_ws;                     // SoA: NGATES x SEQ_LEN (12 MB)
  float* hbuf = gi + (size_t)NGATES * SEQ_LEN;    // SEQ_LEN x 4 (4 MB)

  const int tiles  = SEQ_LEN / 16;   // 16384 wave-tiles
  const int blocks = tiles / 8;      // 8 waves (256 threads) per block

  gi_wmma_kernel<<<blocks, 256, 0, stream>>>(x, w_ih, b_ih, gi);
  gru_scan_kernel<<<1, 32, 0, stream>>>(gi, state, w_hh, b_hh,
                                        hbuf, out + (size_t)SEQ_LEN * 2);
  yproj_wmma_kernel<<<blocks, 256, 0, stream>>>(hbuf, w_out, b_out, out);
}